// MMA_61641370632599
// MI455X (gfx1250) — compile-verified
//
#include <hip/hip_runtime.h>
#include <math.h>

typedef __attribute__((ext_vector_type(2))) float v2f;
typedef __attribute__((ext_vector_type(8))) float v8f;

#define NSA 3.4445f
#define NSB (-4.7750f)
#define NSC 2.0315f
#define BETA 0.8f

__device__ __forceinline__ v8f wmma4(v2f a, v2f b, v8f c) {
    // D = A(16x4,f32) * B(4x16,f32) + C(16x16,f32)
    return __builtin_amdgcn_wmma_f32_16x16x4_f32(false, a, false, b, (short)0, c,
                                                 false, false);
}

__device__ __forceinline__ v8f v8add(v8f a, v8f b) {
    v8f r;
    for (int i = 0; i < 8; ++i) r[i] = a[i] + b[i];
    return r;
}

// ---------------- conv1d(k=3,pad=1) + BN(eval) + ReLU -> h[(b*L+l)*128+d] ----
__global__ void conv_bn_relu_k(const float* __restrict__ x, const float* __restrict__ w,
                               const float* __restrict__ gamma, const float* __restrict__ beta,
                               const float* __restrict__ mean, const float* __restrict__ var,
                               float* __restrict__ h) {
    int idx = blockIdx.x * 256 + threadIdx.x;      // B*L*D = 262144
    int d = idx & 127;
    int l = (idx >> 7) & 255;
    int b = idx >> 15;
    float acc = 0.f;
    for (int c = 0; c < 6; ++c) {
        const float* xr = x + (b * 6 + c) * 256;
        const float* wr = w + (d * 6 + c) * 3;
        if (l > 0)   acc += xr[l - 1] * wr[0];
        acc += xr[l] * wr[1];
        if (l < 255) acc += xr[l + 1] * wr[2];
    }
    float inv = rsqrtf(var[d] + 1e-5f);
    float o = (acc - mean[d]) * inv * gamma[d] + beta[d];
    h[idx] = o > 0.f ? o : 0.f;
}

// ---------------- per-row RMSNorm over d=128 ---------------------------------
__global__ void rmsnorm_k(const float* __restrict__ h, const float* __restrict__ nw,
                          float* __restrict__ hn) {
    int row = blockIdx.x;        // B*L rows
    int d = threadIdx.x;         // 128
    __shared__ float sq[128];
    float v = h[row * 128 + d];
    sq[d] = v * v;
    __syncthreads();
    for (int off = 64; off > 0; off >>= 1) {
        if (d < off) sq[d] += sq[d + off];
        __syncthreads();
    }
    float inv = rsqrtf(sq[0] * (1.f / 128.f) + 1e-6f);
    hn[row * 128 + d] = v * nw[d] * inv;
}

// ---------------- generic WMMA GEMM: C = A(MxK) * B(KxN) [+ Cadd] ------------
// one 16x16 output tile per wave; split-K into 4 accumulators for pipelining
__global__ void wmma_gemm_k(const float* __restrict__ A, const float* __restrict__ B,
                            float* __restrict__ C, const float* __restrict__ Cadd,
                            int Mdim, int Ndim, int Kdim) {
    int lane = threadIdx.x & 31;
    int gw = (blockIdx.x * blockDim.x + threadIdx.x) >> 5;  // global wave id
    int Nt = Ndim >> 4;
    int tiles = (Mdim >> 4) * Nt;
    if (gw >= tiles) return;                 // whole-wave uniform exit
    int m0 = (gw / Nt) << 4;
    int n0 = (gw % Nt) << 4;
    int lm = lane & 15;
    int koff = (lane >> 4) << 1;             // 0 or 2
    int mrow = m0 + lm;
    int ncol = n0 + lm;
    v8f a0 = {}, a1 = {}, a2 = {}, a3 = {};
    for (int kk = 0; kk < Kdim; kk += 16) {  // Kdim is a multiple of 16 here
        v2f a, bf;
#define GEMM_STEP(ACC, KB)                                                     \
        a.x  = A[mrow * Kdim + (KB) + koff];                                   \
        a.y  = A[mrow * Kdim + (KB) + koff + 1];                               \
        bf.x = B[((KB) + koff) * Ndim + ncol];                                 \
        bf.y = B[((KB) + koff + 1) * Ndim + ncol];                             \
        ACC = wmma4(a, bf, ACC);
        GEMM_STEP(a0, kk)
        GEMM_STEP(a1, kk + 4)
        GEMM_STEP(a2, kk + 8)
        GEMM_STEP(a3, kk + 12)
#undef GEMM_STEP
    }
    v8f acc = v8add(v8add(a0, a1), v8add(a2, a3));
    int mb = m0 + ((lane >> 4) << 3);        // M=r (lanes<16) / r+8 (lanes>=16)
    for (int r = 0; r < 8; ++r) {
        float o = acc[r];
        if (Cadd) o += Cadd[(mb + r) * Ndim + ncol];
        C[(mb + r) * Ndim + ncol] = o;
    }
}

// ---------------- momentum selective scan with Muon/Newton-Schulz ------------
// one block (4 waves) per batch element; S,M,X,A,Bm LDS-resident
__global__ __launch_bounds__(128) void scan_k(const float* __restrict__ q,
                                              const float* __restrict__ k,
                                              const float* __restrict__ v,
                                              float* __restrict__ y) {
    const int b = blockIdx.x;
    const int tid = threadIdx.x;     // 0..127, thread owns row d=tid of S/M/X
    const int lane = tid & 31;
    const int wave = tid >> 5;       // 0..3
    __shared__ float Sm[128 * 32];
    __shared__ float Mm[128 * 32];
    __shared__ float Xm[128 * 32];
    __shared__ float Am[32 * 32];
    __shared__ float Bq[32 * 32];
    __shared__ float kt[32], qt[32], vt[128];
    __shared__ float red[4];

    for (int s = 0; s < 32; ++s) { Sm[tid * 32 + s] = 0.f; Mm[tid * 32 + s] = 0.f; }
    // preload t = 0 into LDS
    if (tid < 32) {
        kt[tid] = k[(b * 256) * 32 + tid];
        qt[tid] = q[(b * 256) * 32 + tid];
    }
    vt[tid] = v[(b * 256) * 128 + tid];
    __syncthreads();

    const int si = (wave >> 1) << 4;          // A-tile row origin (0/16)
    const int ti = (wave & 1) << 4;           // A-tile col origin (0/16)
    const int lm = lane & 15;
    const int koff = (lane >> 4) << 1;        // 0 or 2

    for (int t = 0; t < 256; ++t) {
        // issue prefetch of step t+1 (overlaps with the whole NS computation)
        float pk = 0.f, pq = 0.f, pv = 0.f;
        if (t < 255) {
            pv = v[(b * 256 + t + 1) * 128 + tid];
            if (tid < 32) {
                pk = k[(b * 256 + t + 1) * 32 + tid];
                pq = q[(b * 256 + t + 1) * 32 + tid];
            }
        }

        // err = S k_t - v_t ; M = beta*M + err (x) k_t ; accumulate ||M||^2
        float e = -vt[tid];
        for (int s = 0; s < 32; ++s) e += Sm[tid * 32 + s] * kt[s];
        float loc = 0.f;
        for (int s = 0; s < 32; ++s) {
            float m = BETA * Mm[tid * 32 + s] + e * kt[s];
            Mm[tid * 32 + s] = m;
            loc += m * m;
        }
        // wave32 butterfly reduction, then 4-way combine (all threads redundantly)
        loc += __shfl_xor(loc, 16, 32);
        loc += __shfl_xor(loc, 8, 32);
        loc += __shfl_xor(loc, 4, 32);
        loc += __shfl_xor(loc, 2, 32);
        loc += __shfl_xor(loc, 1, 32);
        if (lane == 0) red[wave] = loc;
        __syncthreads();
        float fro = red[0] + red[1] + red[2] + red[3];
        float invn = 1.f / (sqrtf(fro) + 1e-7f);
        for (int s = 0; s < 32; ++s) Xm[tid * 32 + s] = Mm[tid * 32 + s] * invn;
        __syncthreads();

        // ---- 5 Newton-Schulz iterations, all matmuls via WMMA f32 16x16x4 ----
        for (int it = 0; it < 5; ++it) {
            // A = X^T X : output (32x32), K=128. Wave computes tile (si,ti).
            // split-K into 4 independent accumulator chains (depth 8 each)
            v8f c0 = {}, c1 = {}, c2 = {}, c3 = {};
            for (int kk = 0; kk < 128; kk += 16) {
                v2f a, bb;
#define XT_STEP(ACC, KB)                                                       \
                a.x  = Xm[((KB) + koff) * 32 + si + lm];                       \
                a.y  = Xm[((KB) + koff + 1) * 32 + si + lm];                   \
                bb.x = Xm[((KB) + koff) * 32 + ti + lm];                       \
                bb.y = Xm[((KB) + koff + 1) * 32 + ti + lm];                   \
                ACC = wmma4(a, bb, ACC);
                XT_STEP(c0, kk)
                XT_STEP(c1, kk + 4)
                XT_STEP(c2, kk + 8)
                XT_STEP(c3, kk + 12)
#undef XT_STEP
            }
            v8f acc = v8add(v8add(c0, c1), v8add(c2, c3));
            {
                int mb = si + ((lane >> 4) << 3);
                for (int r = 0; r < 8; ++r) Am[(mb + r) * 32 + ti + lm] = acc[r];
            }
            __syncthreads();

            // T = A@A ; Bq = NSB*A + NSC*T  (tile (si,ti) per wave, 2 chains)
            v8f t0 = {}, t1 = {};
            for (int kk = 0; kk < 32; kk += 8) {
                v2f a, bb;
#define AA_STEP(ACC, KB)                                                       \
                a.x  = Am[(si + lm) * 32 + (KB) + koff];                       \
                a.y  = Am[(si + lm) * 32 + (KB) + koff + 1];                   \
                bb.x = Am[((KB) + koff) * 32 + ti + lm];                       \
                bb.y = Am[((KB) + koff + 1) * 32 + ti + lm];                   \
                ACC = wmma4(a, bb, ACC);
                AA_STEP(t0, kk)
                AA_STEP(t1, kk + 4)
#undef AA_STEP
            }
            v8f tt = v8add(t0, t1);
            {
                int mb = si + ((lane >> 4) << 3);
                for (int r = 0; r < 8; ++r) {
                    float av = Am[(mb + r) * 32 + ti + lm];
                    Bq[(mb + r) * 32 + ti + lm] = NSB * av + NSC * tt[r];
                }
            }
            __syncthreads();

            // X = NSA*X + X@Bq : wave owns rows [32*wave, 32*wave+32)
            // 4 independent tiles interleave -> effective chain depth 8
            v8f r0 = {}, r1 = {}, r2 = {}, r3 = {};
            int m0 = wave << 5;
            for (int kk = 0; kk < 32; kk += 4) {
                v2f aT, aB, b0, b1;
                aT.x = Xm[(m0 + lm) * 32 + kk + koff];
                aT.y = Xm[(m0 + lm) * 32 + kk + koff + 1];
                aB.x = Xm[(m0 + 16 + lm) * 32 + kk + koff];
                aB.y = Xm[(m0 + 16 + lm) * 32 + kk + koff + 1];
                b0.x = Bq[(kk + koff) * 32 + lm];
                b0.y = Bq[(kk + koff + 1) * 32 + lm];
                b1.x = Bq[(kk + koff) * 32 + 16 + lm];
                b1.y = Bq[(kk + koff + 1) * 32 + 16 + lm];
                r0 = wmma4(aT, b0, r0);
                r1 = wmma4(aT, b1, r1);
                r2 = wmma4(aB, b0, r2);
                r3 = wmma4(aB, b1, r3);
            }
            int mbT = m0 + ((lane >> 4) << 3);
            int mbB = m0 + 16 + ((lane >> 4) << 3);
            float o0[8], o1[8], o2[8], o3[8];
            for (int r = 0; r < 8; ++r) {     // read old X (all reads precede writes)
                o0[r] = NSA * Xm[(mbT + r) * 32 + lm]      + r0[r];
                o1[r] = NSA * Xm[(mbT + r) * 32 + 16 + lm] + r1[r];
                o2[r] = NSA * Xm[(mbB + r) * 32 + lm]      + r2[r];
                o3[r] = NSA * Xm[(mbB + r) * 32 + 16 + lm] + r3[r];
            }
            for (int r = 0; r < 8; ++r) {
                Xm[(mbT + r) * 32 + lm]      = o0[r];
                Xm[(mbT + r) * 32 + 16 + lm] = o1[r];
                Xm[(mbB + r) * 32 + lm]      = o2[r];
                Xm[(mbB + r) * 32 + 16 + lm] = o3[r];
            }
            __syncthreads();
        }

        // S -= alpha*X (alpha=1) ; y_t = S q_t
        float yv = 0.f;
        for (int s = 0; s < 32; ++s) {
            float sv = Sm[tid * 32 + s] - Xm[tid * 32 + s];
            Sm[tid * 32 + s] = sv;
            yv += sv * qt[s];
        }
        y[(b * 256 + t) * 128 + tid] = yv;
        __syncthreads();                       // everyone done reading kt/qt/vt
        if (t < 255) {                         // commit prefetched step t+1
            vt[tid] = pv;
            if (tid < 32) { kt[tid] = pk; qt[tid] = pq; }
        }
        __syncthreads();
    }
}

// ---------------- mean pool over L + classifier ------------------------------
__global__ void pool_cls_k(const float* __restrict__ h, const float* __restrict__ cw,
                           const float* __restrict__ cb, float* __restrict__ out) {
    __shared__ float pooled[8 * 128];
    int tid = threadIdx.x;                 // 256 threads
    for (int p = 0; p < 4; ++p) {
        int pid = tid + p * 256;           // 0..1023 -> (b,d)
        int b = pid >> 7, d = pid & 127;
        float s = 0.f;
        for (int l = 0; l < 256; ++l) s += h[(b * 256 + l) * 128 + d];
        pooled[pid] = s * (1.f / 256.f);
    }
    __syncthreads();
    int b = tid >> 5, c = tid & 31;
    float s = cb[c];
    for (int d = 0; d < 128; ++d) s += pooled[b * 128 + d] * cw[c * 128 + d];
    out[b * 32 + c] = s;
}

extern "C" void kernel_launch(void* const* d_in, const int* in_sizes, int n_in,
                              void* d_out, int out_size, void* d_ws, size_t ws_size,
                              hipStream_t stream) {
    const float* x        = (const float*)d_in[0];
    const float* conv_w   = (const float*)d_in[1];
    const float* bn_gamma = (const float*)d_in[2];
    const float* bn_beta  = (const float*)d_in[3];
    const float* bn_mean  = (const float*)d_in[4];
    const float* bn_var   = (const float*)d_in[5];
    const float* Wq       = (const float*)d_in[6];
    const float* Wk       = (const float*)d_in[7];
    const float* Wv       = (const float*)d_in[8];
    const float* Wo       = (const float*)d_in[9];
    const float* norm_w   = (const float*)d_in[10];
    const float* cls_w    = (const float*)d_in[11];
    const float* cls_b    = (const float*)d_in[12];
    float* out = (float*)d_out;

    float* ws = (float*)d_ws;
    float* h  = ws;                 // 8*256*128 = 262144
    float* hn = h  + 262144;        // 262144
    float* qb = hn + 262144;        // 65536
    float* kb = qb + 65536;         // 65536
    float* vb = kb + 65536;         // 262144
    float* yb = vb + 262144;        // 262144

    conv_bn_relu_k<<<1024, 256, 0, stream>>>(x, conv_w, bn_gamma, bn_beta,
                                             bn_mean, bn_var, h);
    for (int l = 0; l < 2; ++l) {
        rmsnorm_k<<<2048, 128, 0, stream>>>(h, norm_w + l * 128, hn);
        // q/k: M=2048,N=32,K=128 -> 256 tiles / 8 waves per block = 32 blocks
        wmma_gemm_k<<<32, 256, 0, stream>>>(hn, Wq + l * 128 * 32, qb, nullptr,
                                            2048, 32, 128);
        wmma_gemm_k<<<32, 256, 0, stream>>>(hn, Wk + l * 128 * 32, kb, nullptr,
                                            2048, 32, 128);
        // v: M=2048,N=128,K=128 -> 1024 tiles -> 128 blocks
        wmma_gemm_k<<<128, 256, 0, stream>>>(hn, Wv + l * 128 * 128, vb, nullptr,
                                             2048, 128, 128);
        scan_k<<<8, 128, 0, stream>>>(qb, kb, vb, yb);
        // h = h + y @ Wo
        wmma_gemm_k<<<128, 256, 0, stream>>>(yb, Wo + l * 128 * 128, h, h,
                                             2048, 128, 128);
    }
    pool_cls_k<<<1, 256, 0, stream>>>(h, cls_w, cls_b, out);
}